// InpaintContextAttentionUnit_69578470195753
// MI455X (gfx1250) — compile-verified
//
#include <hip/hip_runtime.h>

typedef __attribute__((ext_vector_type(2))) float v2f;
typedef __attribute__((ext_vector_type(8))) float v8f;

// ---------------- Kernel 1: 32x2 average pool into zero-padded buffer ----------------
// fm: [16,256,256,16]  ->  padded: [16,10,130,16] (interior = pooled, border = 0)
__global__ void pool_pad_kernel(const float* __restrict__ fm, float* __restrict__ padded) {
    int idx = blockIdx.x * blockDim.x + threadIdx.x;
    if (idx >= 16 * 10 * 130) return;
    const int w = idx % 130;
    const int h = (idx / 130) % 10;
    const int b = idx / 1300;

    float4 a0 = make_float4(0.f, 0.f, 0.f, 0.f);
    float4 a1 = a0, a2 = a0, a3 = a0;

    if (h >= 1 && h <= 8 && w >= 1 && w <= 128) {
        const int ph = h - 1, pw2 = w - 1;
        // source block: rows ph*32 .. ph*32+31, cols pw2*2 .. pw2*2+1, all 16 channels
        const float* src = fm + (((b * 256 + ph * 32) * 256) + pw2 * 2) * 16;
        #pragma unroll 4
        for (int i = 0; i < 32; ++i) {
            const float4* p = (const float4*)(src + (size_t)i * 256 * 16);
            // 2 w-positions x 16 channels = 8 float4, contiguous 128B
            a0.x += p[0].x + p[4].x; a0.y += p[0].y + p[4].y; a0.z += p[0].z + p[4].z; a0.w += p[0].w + p[4].w;
            a1.x += p[1].x + p[5].x; a1.y += p[1].y + p[5].y; a1.z += p[1].z + p[5].z; a1.w += p[1].w + p[5].w;
            a2.x += p[2].x + p[6].x; a2.y += p[2].y + p[6].y; a2.z += p[2].z + p[6].z; a2.w += p[2].w + p[6].w;
            a3.x += p[3].x + p[7].x; a3.y += p[3].y + p[7].y; a3.z += p[3].z + p[7].z; a3.w += p[3].w + p[7].w;
        }
        const float s = 1.0f / 64.0f;
        a0.x *= s; a0.y *= s; a0.z *= s; a0.w *= s;
        a1.x *= s; a1.y *= s; a1.z *= s; a1.w *= s;
        a2.x *= s; a2.y *= s; a2.z *= s; a2.w *= s;
        a3.x *= s; a3.y *= s; a3.z *= s; a3.w *= s;
    }
    float4* op = (float4*)(padded + (size_t)idx * 16);
    op[0] = a0; op[1] = a1; op[2] = a2; op[3] = a3;
}

// ---------------- Kernel 2: dual 3x3 conv (+bias +relu) via V_WMMA_F32_16X16X4_F32 ----
// padded: [16,10,130,16]; kern: HWIO [3,3,16,16]; row_op/col_op: [16,8,128,16]
// row conv uses kernel with middle row zeroed -> skip dh==1 taps.
// col conv uses spatial transpose of that    -> skip dw==1 taps (weights at (dw,dh)).
__global__ void conv_wmma_kernel(const float* __restrict__ padded,
                                 const float* __restrict__ kern,
                                 const float* __restrict__ bias,
                                 float* __restrict__ row_op,
                                 float* __restrict__ col_op) {
    const int wave = (blockIdx.x * blockDim.x + threadIdx.x) >> 5;   // 0..1023
    const int lane = threadIdx.x & 31;
    const int hi   = lane >> 4;          // 0/1 : which K-pair / which M-half of C
    const int lm   = lane & 15;          // A: M index (w within tile); B/D: N index (out channel)

    const int wt = wave & 7;             // w-tile (16 wide)
    const int ph = (wave >> 3) & 7;      // pooled row
    const int b  = wave >> 6;            // batch
    const int pw2 = wt * 16 + lm;        // this lane's pooled-w (unpadded coords)

    v8f accR = {0.f, 0.f, 0.f, 0.f, 0.f, 0.f, 0.f, 0.f};
    v8f accC = accR;

    #pragma unroll
    for (int kk = 0; kk < 4; ++kk) {
        const int c0 = kk * 4 + hi * 2;  // this lane's K-pair base (input channel)
        #pragma unroll
        for (int dh = 0; dh < 3; ++dh) {
            #pragma unroll
            for (int dw = 0; dw < 3; ++dw) {
                if (dh == 1 && dw == 1) continue;  // zero in both kernels
                // A[M=pw2(tile-local), K=c0..c0+1] from padded (pad offsets: +dh, +dw)
                const float* ap = padded +
                    (((size_t)(b * 10 + (ph + dh)) * 130) + (pw2 + dw)) * 16 + c0;
                v2f a; a.x = ap[0]; a.y = ap[1];
                if (dh != 1) {  // row-op tap
                    const float* wp = kern + (dh * 3 + dw) * 256 + c0 * 16 + lm;
                    v2f bm; bm.x = wp[0]; bm.y = wp[16];
                    accR = __builtin_amdgcn_wmma_f32_16x16x4_f32(
                        false, a, false, bm, (short)0, accR, false, false);
                }
                if (dw != 1) {  // col-op tap: spatially transposed weights
                    const float* wp = kern + (dw * 3 + dh) * 256 + c0 * 16 + lm;
                    v2f bm; bm.x = wp[0]; bm.y = wp[16];
                    accC = __builtin_amdgcn_wmma_f32_16x16x4_f32(
                        false, a, false, bm, (short)0, accC, false, false);
                }
            }
        }
    }

    const float bv = bias[lm];
    const int base = (b * 8 + ph) * 128 + wt * 16;
    #pragma unroll
    for (int r = 0; r < 8; ++r) {
        const int m = base + r + 8 * hi;      // C/D layout: VGPR r, lanes>=16 hold M=8+r
        float vr = accR[r] + bv; vr = vr > 0.f ? vr : 0.f;
        float vc = accC[r] + bv; vc = vc > 0.f ? vc : 0.f;
        row_op[(size_t)m * 16 + lm] = vr;
        col_op[(size_t)m * 16 + lm] = vc;
    }
}

// ---------------- Kernel 3: bilinear upsample (half-pixel, edge clamp) + fused output --
// One block per (b,h) output row; 256 threads = 256 w positions x 16 channels each.
// Stages the 2 needed source rows of row_op and col_op in LDS.
__global__ void fuse_output_kernel(const float* __restrict__ fm,
                                   const float* __restrict__ row_op,
                                   const float* __restrict__ col_op,
                                   float* __restrict__ out) {
    __shared__ float ldsR[2][128 * 16];
    __shared__ float ldsC[2][128 * 16];

    const int h = blockIdx.x & 255;
    const int b = blockIdx.x >> 8;

    // y mapping: src_y = (h+0.5)*(8/256) - 0.5
    const float srcy = (h + 0.5f) * (1.0f / 32.0f) - 0.5f;
    const float yf = floorf(srcy);
    const float fy = srcy - yf;
    int y0 = (int)yf, y1 = y0 + 1;
    y0 = y0 < 0 ? 0 : (y0 > 7 ? 7 : y0);
    y1 = y1 < 0 ? 0 : (y1 > 7 ? 7 : y1);

    const float4* rb0 = (const float4*)(row_op + (size_t)(b * 8 + y0) * 2048);
    const float4* rb1 = (const float4*)(row_op + (size_t)(b * 8 + y1) * 2048);
    const float4* cb0 = (const float4*)(col_op + (size_t)(b * 8 + y0) * 2048);
    const float4* cb1 = (const float4*)(col_op + (size_t)(b * 8 + y1) * 2048);
    #pragma unroll
    for (int i = 0; i < 2; ++i) {
        const int j = threadIdx.x + i * 256;   // 512 float4 per plane
        ((float4*)ldsR[0])[j] = rb0[j];
        ((float4*)ldsR[1])[j] = rb1[j];
        ((float4*)ldsC[0])[j] = cb0[j];
        ((float4*)ldsC[1])[j] = cb1[j];
    }
    __syncthreads();

    const int w = threadIdx.x;
    // x mapping: src_x = (w+0.5)*0.5 - 0.5
    const float srcx = 0.5f * w - 0.25f;
    const float xf = floorf(srcx);
    const float fx = srcx - xf;
    int x0 = (int)xf, x1 = x0 + 1;
    x0 = x0 < 0 ? 0 : x0;
    x1 = x1 > 127 ? 127 : x1;

    const float w00 = (1.f - fy) * (1.f - fx), w01 = (1.f - fy) * fx;
    const float w10 = fy * (1.f - fx),         w11 = fy * fx;

    const float4* fmp = (const float4*)(fm + ((size_t)(b * 256 + h) * 256 + w) * 16);
    float* op = out + ((size_t)(b * 256 + h) * 256 + w) * 48;

    #pragma unroll
    for (int c4 = 0; c4 < 4; ++c4) {
        const float4 f = fmp[c4];
        const float4 r00 = *(const float4*)&ldsR[0][x0 * 16 + c4 * 4];
        const float4 r01 = *(const float4*)&ldsR[0][x1 * 16 + c4 * 4];
        const float4 r10 = *(const float4*)&ldsR[1][x0 * 16 + c4 * 4];
        const float4 r11 = *(const float4*)&ldsR[1][x1 * 16 + c4 * 4];
        const float4 c00 = *(const float4*)&ldsC[0][x0 * 16 + c4 * 4];
        const float4 c01 = *(const float4*)&ldsC[0][x1 * 16 + c4 * 4];
        const float4 c10 = *(const float4*)&ldsC[1][x0 * 16 + c4 * 4];
        const float4 c11 = *(const float4*)&ldsC[1][x1 * 16 + c4 * 4];

        float4 ru, cu, dR, dC;
        ru.x = w00 * r00.x + w01 * r01.x + w10 * r10.x + w11 * r11.x;
        ru.y = w00 * r00.y + w01 * r01.y + w10 * r10.y + w11 * r11.y;
        ru.z = w00 * r00.z + w01 * r01.z + w10 * r10.z + w11 * r11.z;
        ru.w = w00 * r00.w + w01 * r01.w + w10 * r10.w + w11 * r11.w;
        cu.x = w00 * c00.x + w01 * c01.x + w10 * c10.x + w11 * c11.x;
        cu.y = w00 * c00.y + w01 * c01.y + w10 * c10.y + w11 * c11.y;
        cu.z = w00 * c00.z + w01 * c01.z + w10 * c10.z + w11 * c11.z;
        cu.w = w00 * c00.w + w01 * c01.w + w10 * c10.w + w11 * c11.w;

        dR.x = f.x - ru.x; dR.y = f.y - ru.y; dR.z = f.z - ru.z; dR.w = f.w - ru.w;
        dC.x = f.x - cu.x; dC.y = f.y - cu.y; dC.z = f.z - cu.z; dC.w = f.w - cu.w;

        ((float4*)(op))[c4]      = f;   // channels [0,16)
        ((float4*)(op + 16))[c4] = dR;  // channels [16,32)
        ((float4*)(op + 32))[c4] = dC;  // channels [32,48)
    }
}

extern "C" void kernel_launch(void* const* d_in, const int* in_sizes, int n_in,
                              void* d_out, int out_size, void* d_ws, size_t ws_size,
                              hipStream_t stream) {
    const float* fm   = (const float*)d_in[0];  // [16,256,256,16]
    const float* kern = (const float*)d_in[1];  // [3,3,16,16]
    const float* bias = (const float*)d_in[2];  // [16]
    float* out = (float*)d_out;                 // [16,256,256,48]
    float* ws  = (float*)d_ws;

    float* padded = ws;                         // 16*10*130*16 = 332800 floats
    float* row_op = ws + 332800;                // 16*8*128*16  = 262144 floats
    float* col_op = ws + 332800 + 262144;       // 262144 floats   (total ~3.4 MB)

    pool_pad_kernel<<<(16 * 10 * 130 + 255) / 256, 256, 0, stream>>>(fm, padded);
    conv_wmma_kernel<<<128, 256, 0, stream>>>(padded, kern, bias, row_op, col_op);
    fuse_output_kernel<<<16 * 256, 256, 0, stream>>>(fm, row_op, col_op, out);
}